// NGCF_71657234366495
// MI455X (gfx1250) — compile-verified
//
#include <hip/hip_runtime.h>

#define N_NODES 100000
#define N_EDGES_C 3200000
#define EMB 128
#define LDSTRIDE 132   // pad 16x128 fp32 tile rows to avoid LDS bank conflicts

typedef __attribute__((ext_vector_type(2))) float v2f;
typedef __attribute__((ext_vector_type(8))) float v8f;

// ---------------------------------------------------------------------------
// Prologue: mean = emb/4, section1 = emb copy
// ---------------------------------------------------------------------------
__global__ void ngcf_init(const float* __restrict__ emb,
                          float* __restrict__ mean_out,
                          float* __restrict__ emb_copy, size_t n) {
  size_t i = (size_t)blockIdx.x * blockDim.x + threadIdx.x;
  if (i < n) {
    float e = emb[i];
    mean_out[i] = 0.25f * e;
    emb_copy[i] = e;
  }
}

__global__ void ngcf_zero(float* __restrict__ x, size_t n) {
  size_t i = (size_t)blockIdx.x * blockDim.x + threadIdx.x;
  if (i < n) x[i] = 0.0f;
}

// ---------------------------------------------------------------------------
// COO SpMM: x[row] += val * src[col]   (one wave32 per edge, float4 per lane)
// ---------------------------------------------------------------------------
__global__ __launch_bounds__(256) void ngcf_spmm(
    const int* __restrict__ rows, const int* __restrict__ cols,
    const float* __restrict__ ev, const float* __restrict__ src,
    float* __restrict__ x) {
  int e = (int)((blockIdx.x * blockDim.x + threadIdx.x) >> 5);
  int lane = threadIdx.x & 31;
  if (e >= N_EDGES_C) return;
  int r = rows[e];
  int c = cols[e];
  float v = ev[e];
  const float4* s = (const float4*)(src + (size_t)c * EMB);
  float4 f = s[lane];                       // 32 lanes x 4 floats = 128
  float* dst = x + (size_t)r * EMB + lane * 4;
  atomicAdd(dst + 0, v * f.x);              // global_atomic_add_f32
  atomicAdd(dst + 1, v * f.y);
  atomicAdd(dst + 2, v * f.z);
  atomicAdd(dst + 3, v * f.w);
}

// ---------------------------------------------------------------------------
// Fused dual-GEMM + bias + LeakyReLU + row L2 norm + mean accumulation.
// One workgroup (8 waves) handles 16 rows x 128 cols; wave w owns cols
// [16w, 16w+16). fp32 WMMA 16x16x4 keeps full reference precision.
// ---------------------------------------------------------------------------
__global__ __launch_bounds__(256) void ngcf_gemm_norm(
    const float* __restrict__ ein, const float* __restrict__ x,
    const float* __restrict__ W1, const float* __restrict__ b1,
    const float* __restrict__ W2, const float* __restrict__ b2,
    float* __restrict__ hout, float* __restrict__ mean_out) {
  __shared__ float A1[16 * LDSTRIDE];   // emb + x
  __shared__ float A2[16 * LDSTRIDE];   // x * emb
  __shared__ float rowSq[16];

  const int t = threadIdx.x;
  const int rb = blockIdx.x * 16;       // 100000 / 16 = 6250 blocks, exact

  // Stage A tiles: 16 threads per row, 2 float4 each.
  {
    int row = t >> 4;
    int c4 = t & 15;
    const float4* ep = (const float4*)(ein + (size_t)(rb + row) * EMB);
    const float4* xp = (const float4*)(x + (size_t)(rb + row) * EMB);
#pragma unroll
    for (int h = 0; h < 2; ++h) {
      int c = c4 + 16 * h;
      float4 e = ep[c];
      float4 xv = xp[c];
      float* a1 = A1 + row * LDSTRIDE + c * 4;
      float* a2 = A2 + row * LDSTRIDE + c * 4;
      a1[0] = e.x + xv.x; a1[1] = e.y + xv.y;
      a1[2] = e.z + xv.z; a1[3] = e.w + xv.w;
      a2[0] = e.x * xv.x; a2[1] = e.y * xv.y;
      a2[2] = e.z * xv.z; a2[3] = e.w * xv.w;
    }
  }
  if (t < 16) rowSq[t] = 0.0f;
  __syncthreads();

  const int wave = t >> 5;              // column tile 0..7
  const int lane = t & 31;
  const int m = lane & 15;              // A row / D column-within-tile
  const int kh = lane >> 4;             // K-half select (K= 2*kh, 2*kh+1)
  const int ncol = wave * 16 + m;       // global output column

  const float* a1row = A1 + m * LDSTRIDE + 2 * kh;
  const float* a2row = A2 + m * LDSTRIDE + 2 * kh;

  v8f acc = {};
#pragma unroll 4
  for (int k0 = 0; k0 < EMB; k0 += 4) {
    // A fragment (16x4 fp32): lane holds A[m][k0+2kh], A[m][k0+2kh+1]
    v2f af1; af1.x = a1row[k0]; af1.y = a1row[k0 + 1];
    // B fragment (4x16 fp32): lane holds W[k0+2kh][ncol], W[k0+2kh+1][ncol]
    v2f bf1;
    bf1.x = W1[(size_t)(k0 + 2 * kh) * EMB + ncol];
    bf1.y = W1[(size_t)(k0 + 2 * kh + 1) * EMB + ncol];
    acc = __builtin_amdgcn_wmma_f32_16x16x4_f32(false, af1, false, bf1,
                                                (short)0, acc, false, false);

    v2f af2; af2.x = a2row[k0]; af2.y = a2row[k0 + 1];
    v2f bf2;
    bf2.x = W2[(size_t)(k0 + 2 * kh) * EMB + ncol];
    bf2.y = W2[(size_t)(k0 + 2 * kh + 1) * EMB + ncol];
    acc = __builtin_amdgcn_wmma_f32_16x16x4_f32(false, af2, false, bf2,
                                                (short)0, acc, false, false);
  }

  // D layout: acc[r] -> row (r + 8*kh), col ncol
  const float bias = b1[ncol] + b2[ncol];
  float vals[8];
#pragma unroll
  for (int r = 0; r < 8; ++r) {
    float hv = acc[r] + bias;
    hv = (hv >= 0.0f) ? hv : 0.2f * hv;          // LeakyReLU(0.2)
    vals[r] = hv;
    atomicAdd(&rowSq[r + 8 * kh], hv * hv);      // ds_add_f32
  }
  __syncthreads();

#pragma unroll
  for (int r = 0; r < 8; ++r) {
    int row = r + 8 * kh;
    float nrm = sqrtf(rowSq[row]);
    float scale = 1.0f / fmaxf(nrm, 1e-12f);
    float hv = vals[r] * scale;
    size_t idx = (size_t)(rb + row) * EMB + ncol;
    hout[idx] = hv;
    mean_out[idx] += 0.25f * hv;                 // sequential kernels: safe RMW
  }
}

// ---------------------------------------------------------------------------
extern "C" void kernel_launch(void* const* d_in, const int* in_sizes, int n_in,
                              void* d_out, int out_size, void* d_ws,
                              size_t ws_size, hipStream_t stream) {
  const float* emb = (const float*)d_in[0];
  const int* ei    = (const int*)d_in[1];   // [2, N_EDGES]: rows then cols
  const float* ev  = (const float*)d_in[2];
  const float* W1  = (const float*)d_in[3]; // [3,128,128]
  const float* b1  = (const float*)d_in[4]; // [3,128]
  const float* W2  = (const float*)d_in[5];
  const float* b2  = (const float*)d_in[6];
  float* out = (float*)d_out;               // [mean | emb | h1 | h2 | h3]
  float* x = (float*)d_ws;                  // [N_NODES, EMB] scratch

  const size_t ND = (size_t)N_NODES * EMB;
  float* mean = out;

  ngcf_init<<<(unsigned)((ND + 255) / 256), 256, 0, stream>>>(emb, mean,
                                                              out + ND, ND);

  for (int l = 0; l < 3; ++l) {
    ngcf_zero<<<(unsigned)((ND + 255) / 256), 256, 0, stream>>>(x, ND);

    const float* src = out + (size_t)(1 + l) * ND;  // previous layer embedding
    ngcf_spmm<<<N_EDGES_C / 8, 256, 0, stream>>>(ei, ei + N_EDGES_C, ev, src, x);

    ngcf_gemm_norm<<<N_NODES / 16, 256, 0, stream>>>(
        src, x,
        W1 + (size_t)l * EMB * EMB, b1 + (size_t)l * EMB,
        W2 + (size_t)l * EMB * EMB, b2 + (size_t)l * EMB,
        out + (size_t)(2 + l) * ND, mean);
  }
}